// GNNModel_82987358094100
// MI455X (gfx1250) — compile-verified
//
#include <hip/hip_runtime.h>
#include <stdint.h>

#define NEG_ATT 0.2f
#define NEG_ACT 0.01f
#define BN_EPS  1e-5f
#define TPB     256
#define TILE    (2 * TPB)   // 512 edges staged per block-iteration

// ---- float atomic max via int ordering trick (valid for any sign mix) ----
__device__ __forceinline__ void atomic_max_f32(float* addr, float val) {
  if (val >= 0.0f) atomicMax((int*)addr, __float_as_int(val));
  else             atomicMin((unsigned int*)addr, __float_as_uint(val));
}

// ---- CDNA5 async global->LDS copies (gfx1250), GVS addressing mode ----
// LDS dest at lds_byte ; global src at sbase + voff_bytes (8 bytes -> 2 edges/lane)
__device__ __forceinline__ void async_ld_b64(unsigned lds_byte, unsigned voff_bytes,
                                             unsigned long long sbase) {
  asm volatile("global_load_async_to_lds_b64 %0, %1, %2"
               :: "v"(lds_byte), "v"(voff_bytes), "s"(sbase) : "memory");
}
__device__ __forceinline__ void wait_async0() {
  asm volatile("s_wait_asynccnt 0x0" ::: "memory");
}
__device__ __forceinline__ void wait_async2() {   // previous tile (2 older ops) retired
  asm volatile("s_wait_asynccnt 0x2" ::: "memory");
}

// =====================  edge kernels (pipelined async-LDS staged index stream)  ============
// fused softmax-sum: e = exp(alpha - m[dst]); denom[dst] += e; numer[dst] += h[src]*e
__global__ void edge_sum_kernel(const int* __restrict__ ei, int Etot,
                                const float* __restrict__ h,
                                const float* __restrict__ a_s,
                                const float* __restrict__ a_d,
                                const float* __restrict__ m,
                                float* __restrict__ denom,
                                float* __restrict__ numer) {
  __shared__ __align__(16) int s_src[2][TILE];
  __shared__ __align__(16) int s_dst[2][TILE];
  const int t = threadIdx.x;
  const unsigned long long srcb = (unsigned long long)(size_t)ei;
  const unsigned long long dstb = (unsigned long long)(size_t)(ei + Etot);
  const unsigned l_s0 = (unsigned)(size_t)&s_src[0][2 * t];
  const unsigned l_d0 = (unsigned)(size_t)&s_dst[0][2 * t];
  const unsigned l_s1 = (unsigned)(size_t)&s_src[1][2 * t];
  const unsigned l_d1 = (unsigned)(size_t)&s_dst[1][2 * t];
  const int nstep = gridDim.x * TILE;
  int cur = blockIdx.x * TILE;
  int parity = 0;

  if (cur + TILE <= Etot) {            // prologue: stage first tile
    unsigned boff = (unsigned)(cur + 2 * t) * 4u;
    async_ld_b64(l_s0, boff, srcb);
    async_ld_b64(l_d0, boff, dstb);
  }
  for (; cur + TILE <= Etot; cur += nstep, parity ^= 1) {
    const int nxt = cur + nstep;
    if (nxt + TILE <= Etot) {          // stage next tile into other buffer
      unsigned boff = (unsigned)(nxt + 2 * t) * 4u;
      async_ld_b64(parity ? l_s0 : l_s1, boff, srcb);
      async_ld_b64(parity ? l_d0 : l_d1, boff, dstb);
      if ((t & 7) == 0) {              // prefetch two strides ahead (covers 2KB/row tile)
        int pf = cur + 2 * nstep + 2 * t;
        if (pf + TILE <= Etot) {
          __builtin_prefetch(ei + pf, 0, 0);
          __builtin_prefetch(ei + Etot + pf, 0, 0);
        }
      }
      wait_async2();                   // current tile's 2 ops retired, next still in flight
    } else {
      wait_async0();
    }
    #pragma unroll
    for (int j = 0; j < 2; ++j) {
      int s = s_src[parity][2 * t + j];
      int d = s_dst[parity][2 * t + j];
      float al = a_s[s] + a_d[d];
      al = (al > 0.0f) ? al : NEG_ATT * al;
      float ev = __expf(al - m[d]);
      atomicAdd(&denom[d], ev);
      atomicAdd(&numer[d], ev * h[s]);
    }
  }
  // tail (< TILE edges): direct loads
  const int tailStart = (Etot / TILE) * TILE;
  for (int e = tailStart + blockIdx.x * blockDim.x + t; e < Etot;
       e += gridDim.x * blockDim.x) {
    int s = ei[e];
    int d = ei[Etot + e];
    float al = a_s[s] + a_d[d];
    al = (al > 0.0f) ? al : NEG_ATT * al;
    float ev = __expf(al - m[d]);
    atomicAdd(&denom[d], ev);
    atomicAdd(&numer[d], ev * h[s]);
  }
}

// segment max of leaky-relu attention logits over incoming edges
__global__ void edge_max_kernel(const int* __restrict__ ei, int Etot,
                                const float* __restrict__ a_s,
                                const float* __restrict__ a_d,
                                float* __restrict__ m) {
  __shared__ __align__(16) int s_src[2][TILE];
  __shared__ __align__(16) int s_dst[2][TILE];
  const int t = threadIdx.x;
  const unsigned long long srcb = (unsigned long long)(size_t)ei;
  const unsigned long long dstb = (unsigned long long)(size_t)(ei + Etot);
  const unsigned l_s0 = (unsigned)(size_t)&s_src[0][2 * t];
  const unsigned l_d0 = (unsigned)(size_t)&s_dst[0][2 * t];
  const unsigned l_s1 = (unsigned)(size_t)&s_src[1][2 * t];
  const unsigned l_d1 = (unsigned)(size_t)&s_dst[1][2 * t];
  const int nstep = gridDim.x * TILE;
  int cur = blockIdx.x * TILE;
  int parity = 0;

  if (cur + TILE <= Etot) {
    unsigned boff = (unsigned)(cur + 2 * t) * 4u;
    async_ld_b64(l_s0, boff, srcb);
    async_ld_b64(l_d0, boff, dstb);
  }
  for (; cur + TILE <= Etot; cur += nstep, parity ^= 1) {
    const int nxt = cur + nstep;
    if (nxt + TILE <= Etot) {
      unsigned boff = (unsigned)(nxt + 2 * t) * 4u;
      async_ld_b64(parity ? l_s0 : l_s1, boff, srcb);
      async_ld_b64(parity ? l_d0 : l_d1, boff, dstb);
      if ((t & 7) == 0) {
        int pf = cur + 2 * nstep + 2 * t;
        if (pf + TILE <= Etot) {
          __builtin_prefetch(ei + pf, 0, 0);
          __builtin_prefetch(ei + Etot + pf, 0, 0);
        }
      }
      wait_async2();
    } else {
      wait_async0();
    }
    #pragma unroll
    for (int j = 0; j < 2; ++j) {
      int s = s_src[parity][2 * t + j];
      int d = s_dst[parity][2 * t + j];
      float al = a_s[s] + a_d[d];
      al = (al > 0.0f) ? al : NEG_ATT * al;
      atomic_max_f32(&m[d], al);
    }
  }
  const int tailStart = (Etot / TILE) * TILE;
  for (int e = tailStart + blockIdx.x * blockDim.x + t; e < Etot;
       e += gridDim.x * blockDim.x) {
    int s = ei[e];
    int d = ei[Etot + e];
    float al = a_s[s] + a_d[d];
    al = (al > 0.0f) ? al : NEG_ATT * al;
    atomic_max_f32(&m[d], al);
  }
}

// =====================  node kernels  =====================
__global__ void copy_kernel(const float* __restrict__ src, float* __restrict__ dst, int N) {
  int n = blockIdx.x * blockDim.x + threadIdx.x;
  if (n < N) dst[n] = src[n];
}

// h = x*W ; a_src = h*att_src ; a_dst = h*att_dst ; m = leakyrelu(a_src+a_dst) (self-loop)
__global__ void node_prep_kernel(const float* __restrict__ x_cur,
                                 const float* __restrict__ W,
                                 const float* __restrict__ att_s,
                                 const float* __restrict__ att_d,
                                 float* __restrict__ h, float* __restrict__ a_s,
                                 float* __restrict__ a_d, float* __restrict__ m, int N) {
  int n = blockIdx.x * blockDim.x + threadIdx.x;
  if (n >= N) return;
  float hv = x_cur[n] * W[0];
  float av = hv * att_s[0];
  float dv = hv * att_d[0];
  h[n] = hv; a_s[n] = av; a_d[n] = dv;
  float al = av + dv;
  m[n] = (al > 0.0f) ? al : NEG_ATT * al;   // self-loop always present -> valid max init
}

// denom/numer initialized with the analytic self-loop term; thread 0 zeros BN stats
__global__ void node_selfinit_kernel(const float* __restrict__ h,
                                     const float* __restrict__ a_s,
                                     const float* __restrict__ a_d,
                                     const float* __restrict__ m,
                                     float* __restrict__ denom, float* __restrict__ numer,
                                     float* __restrict__ stats, int N) {
  int n = blockIdx.x * blockDim.x + threadIdx.x;
  if (n == 0) { stats[0] = 0.0f; stats[1] = 0.0f; }
  if (n >= N) return;
  float al = a_s[n] + a_d[n];
  al = (al > 0.0f) ? al : NEG_ATT * al;
  float ev = __expf(al - m[n]);
  denom[n] = ev;
  numer[n] = ev * h[n];
}

// hout = numer/(denom+1e-16) + bias ; block-reduce sum & sumsq into stats
__global__ void node_hout_stats_kernel(const float* __restrict__ numer,
                                       const float* __restrict__ denom,
                                       const float* __restrict__ bias,
                                       float* __restrict__ hout,
                                       float* __restrict__ stats, int N) {
  __shared__ float ss[TPB];
  __shared__ float sq[TPB];
  int t = threadIdx.x;
  int n = blockIdx.x * blockDim.x + t;
  float hv = 0.0f;
  if (n < N) {
    hv = numer[n] / (denom[n] + 1e-16f) + bias[0];
    hout[n] = hv;
  }
  ss[t] = (n < N) ? hv : 0.0f;
  sq[t] = (n < N) ? hv * hv : 0.0f;
  __syncthreads();
  for (int off = TPB / 2; off > 0; off >>= 1) {
    if (t < off) { ss[t] += ss[t + off]; sq[t] += sq[t + off]; }
    __syncthreads();
  }
  if (t == 0) { atomicAdd(&stats[0], ss[0]); atomicAdd(&stats[1], sq[0]); }
}

// BatchNorm (training, biased var) + LeakyReLU(0.01) + residual (original x)
__global__ void node_bn_act_kernel(const float* __restrict__ hout,
                                   const float* __restrict__ stats,
                                   const float* __restrict__ gamma,
                                   const float* __restrict__ beta,
                                   const float* __restrict__ x_res,
                                   float* __restrict__ x_cur, int N, float invN) {
  int n = blockIdx.x * blockDim.x + threadIdx.x;
  if (n >= N) return;
  float mu  = stats[0] * invN;
  float var = stats[1] * invN - mu * mu;
  float rs  = rsqrtf(fmaxf(var, 0.0f) + BN_EPS);
  float hv  = gamma[0] * (hout[n] - mu) * rs + beta[0];
  hv = (hv > 0.0f) ? hv : NEG_ACT * hv;
  x_cur[n] = hv + x_res[n];
}

// =====================  pooling  =====================
__global__ void zero_pool_kernel(float* __restrict__ pool, int B) {
  int t = threadIdx.x;
  if (t < B) pool[t] = 0.0f;
}

// masked per-graph sum; each 256-thread block straddles at most 2 graphs -> 2-bin reduce
__global__ void pool_kernel(const float* __restrict__ x_cur,
                            const float* __restrict__ dec,
                            float* __restrict__ pool, int N, int NPG, int B) {
  __shared__ float s0[TPB];
  __shared__ float s1[TPB];
  int t = threadIdx.x;
  int n = blockIdx.x * blockDim.x + t;
  int bFirst = (blockIdx.x * blockDim.x) / NPG;
  float v0 = 0.0f, v1 = 0.0f;
  if (n < N) {
    int b = n / NPG;
    int r = n - b * NPG;
    float v = x_cur[n] * dec[r];
    if (b == bFirst) v0 = v; else v1 = v;
  }
  s0[t] = v0; s1[t] = v1;
  __syncthreads();
  for (int off = TPB / 2; off > 0; off >>= 1) {
    if (t < off) { s0[t] += s0[t + off]; s1[t] += s1[t + off]; }
    __syncthreads();
  }
  if (t == 0) {
    atomicAdd(&pool[bFirst], s0[0]);
    if (bFirst + 1 < B) atomicAdd(&pool[bFirst + 1], s1[0]);
  }
}

__global__ void finalize_kernel(const float* __restrict__ pool, float* __restrict__ out,
                                int B, float invNPG) {
  __shared__ float vals[130];
  int t = threadIdx.x;
  float v = 0.0f;
  if (t < B) { v = pool[t] * invNPG; vals[t] = v; }
  __syncthreads();
  if (t == 0) {
    float mn = vals[0], mx = vals[0];
    for (int i = 1; i < B; ++i) { mn = fminf(mn, vals[i]); mx = fmaxf(mx, vals[i]); }
    vals[128] = mn; vals[129] = mx;
  }
  __syncthreads();
  if (t < B) out[t] = (v - vals[128]) / (vals[129] - vals[128] + 1e-6f);
}

// =====================  host orchestration  =====================
extern "C" void kernel_launch(void* const* d_in, const int* in_sizes, int n_in,
                              void* d_out, int out_size, void* d_ws, size_t ws_size,
                              hipStream_t stream) {
  const float* x     = (const float*)d_in[0];
  const int*   ei    = (const int*)  d_in[1];
  const float* W     = (const float*)d_in[3];
  const float* att_s = (const float*)d_in[4];
  const float* att_d = (const float*)d_in[5];
  const float* bias  = (const float*)d_in[6];
  const float* gamma = (const float*)d_in[7];
  const float* beta  = (const float*)d_in[8];
  const float* dec   = (const float*)d_in[9];

  const int N    = in_sizes[0];       // 800000 nodes
  const int Etot = in_sizes[1] / 2;   // 25.6M edges
  const int NPG  = in_sizes[9];       // 50000 nodes per graph
  const int B    = out_size;          // 16 graphs

  float* ws    = (float*)d_ws;
  float* x_cur = ws + 0LL * N;
  float* h     = ws + 1LL * N;
  float* a_s   = ws + 2LL * N;
  float* a_d   = ws + 3LL * N;
  float* m     = ws + 4LL * N;  // reused as hout after edge passes
  float* denom = ws + 5LL * N;
  float* numer = ws + 6LL * N;
  float* stats = ws + 7LL * N;  // [sum, sumsq]
  float* pool  = stats + 8;

  dim3 blk(TPB);
  const int nodeGrid = (N + TPB - 1) / TPB;
  const int edgeGrid = 8192;
  const float invN = 1.0f / (float)N;

  copy_kernel<<<nodeGrid, blk, 0, stream>>>(x, x_cur, N);

  for (int p = 0; p < 4; ++p) {
    node_prep_kernel<<<nodeGrid, blk, 0, stream>>>(x_cur, W, att_s, att_d, h, a_s, a_d, m, N);
    edge_max_kernel<<<edgeGrid, blk, 0, stream>>>(ei, Etot, a_s, a_d, m);
    node_selfinit_kernel<<<nodeGrid, blk, 0, stream>>>(h, a_s, a_d, m, denom, numer, stats, N);
    edge_sum_kernel<<<edgeGrid, blk, 0, stream>>>(ei, Etot, h, a_s, a_d, m, denom, numer);
    node_hout_stats_kernel<<<nodeGrid, blk, 0, stream>>>(numer, denom, bias, m, stats, N);
    node_bn_act_kernel<<<nodeGrid, blk, 0, stream>>>(m, stats, gamma, beta, x, x_cur, N, invN);
  }

  zero_pool_kernel<<<1, 64, 0, stream>>>(pool, B);
  pool_kernel<<<nodeGrid, blk, 0, stream>>>(x_cur, dec, pool, N, NPG, B);
  finalize_kernel<<<1, 64, 0, stream>>>(pool, (float*)d_out, B, 1.0f / (float)NPG);
}